// MedianPool2d_10668698763698
// MI455X (gfx1250) — compile-verified
//
#include <hip/hip_runtime.h>
#include <hip/hip_bf16.h>

// Median pool 3x3, stride 1, no padding.
// in : [16,3,1024,1024] fp32   out : [16,3,1022,1022] fp32
// Bandwidth-bound (~400 MB -> ~17us floor @ 23.3 TB/s).
// Strategy: stage 130x18 input tiles in LDS via gfx1250 async global->LDS DMA
// (ASYNCcnt + s_wait_asynccnt), compute medians with the column-sort
// decomposition (pure v_min/max_num_f32), stream results out with NT stores
// so the 192MB L2 stays dedicated to input-halo reuse.

#define IH 1024
#define IW 1024
#define HO 1022
#define WO 1022
#define TILE_X 128
#define TILE_Y 16
#define PITCH  132            // LDS row pitch (floats), 16B-aligned rows
#define LROWS  18             // TILE_Y + 2
#define NSEG   33             // float4 segments per LDS row (132 floats)
#define NLOAD  (LROWS * NSEG) // 594 async b128 transfers per tile

typedef float v2f __attribute__((ext_vector_type(2)));  // native clang vector

__device__ __forceinline__ float med3f(float a, float b, float c) {
    return fmaxf(fminf(a, b), fminf(fmaxf(a, b), c));
}

// Store one 4-wide output row segment with non-temporal hints.
// rowp is 8B-aligned (ox % 4 == 0, row stride 1022*4 ≡ 0 mod 8).
__device__ __forceinline__ void store_row_nt(float* __restrict__ rowp,
                                             const float r[4], int ox) {
    if (ox + 3 < WO) {
        v2f p0; p0.x = r[0]; p0.y = r[1];
        v2f p1; p1.x = r[2]; p1.y = r[3];
        __builtin_nontemporal_store(p0, (v2f*)rowp);
        __builtin_nontemporal_store(p1, (v2f*)(rowp + 2));
    } else {
        if (ox     < WO) __builtin_nontemporal_store(r[0], rowp);
        if (ox + 1 < WO) __builtin_nontemporal_store(r[1], rowp + 1);
        if (ox + 2 < WO) __builtin_nontemporal_store(r[2], rowp + 2);
    }
}

__global__ __launch_bounds__(256) void median3x3_kernel(
        const float* __restrict__ in, float* __restrict__ out) {
    __shared__ __align__(16) float tile[LROWS * PITCH];

    const int tid = threadIdx.x;
    const int x0  = blockIdx.x * TILE_X;
    const int y0  = blockIdx.y * TILE_Y;
    const int img = blockIdx.z;
    // 48 * 1024 * 1024 = 50.3M elements < 2^31: 32-bit offsets are safe.
    const float* __restrict__ src = in + (unsigned)img * (unsigned)(IH * IW);

    // Low 32 bits of the flat pointer == wave-relative LDS byte offset.
    const unsigned lds_base = (unsigned)(unsigned long long)(uintptr_t)&tile[0];

    // ---- Stage tile: async global -> LDS (16B per lane per op) ----
    #pragma unroll
    for (int k = 0; k < 3; ++k) {
        const int idx = tid + k * 256;
        if (idx < NLOAD) {
            const int r = idx / NSEG;
            const int s = idx - r * NSEG;
            int gy = y0 + r;      gy = gy > (IH - 1) ? (IH - 1) : gy;  // clamp rows
            int gx = x0 + s * 4;  gx = gx > (IW - 4) ? (IW - 4) : gx;  // keep b128 in-row
            const unsigned long long gaddr =
                (unsigned long long)(src + (unsigned)(gy * IW + gx));
            const unsigned laddr = lds_base + (unsigned)((r * PITCH + s * 4) * 4);
            asm volatile("global_load_async_to_lds_b128 %0, %1, off"
                         :: "v"(laddr), "v"(gaddr) : "memory");
        }
    }
    asm volatile("s_wait_asynccnt 0" ::: "memory"); // this wave's DMAs complete
    __syncthreads();                                 // all waves' DMAs visible

    // ---- Compute: each thread -> 4 wide x 2 tall outputs ----
    const int tx = tid & 31;
    const int ty = tid >> 5;
    const int lx = tx * 4;
    const int ly = ty * 2;
    const int ox = x0 + lx;
    const int oy = y0 + ly;

    // 4 input rows x 6 input cols from LDS (ds_load_b128 + ds_load_b64 per row)
    float v[4][6];
    #pragma unroll
    for (int r = 0; r < 4; ++r) {
        const float* rp = &tile[(ly + r) * PITCH + lx];
        const float4 a = *(const float4*)rp;
        const float2 b = *(const float2*)(rp + 4);
        v[r][0] = a.x; v[r][1] = a.y; v[r][2] = a.z; v[r][3] = a.w;
        v[r][4] = b.x; v[r][5] = b.y;
    }

    // Sort each 3-element column for both output rows, sharing minmax(v1,v2).
    float loA[6], miA[6], hiA[6];   // output row oy   (input rows 0,1,2)
    float loB[6], miB[6], hiB[6];   // output row oy+1 (input rows 1,2,3)
    #pragma unroll
    for (int c = 0; c < 6; ++c) {
        const float a = fminf(v[1][c], v[2][c]);
        const float b = fmaxf(v[1][c], v[2][c]);
        float t = v[0][c];
        loA[c] = fminf(t, a);
        hiA[c] = fmaxf(t, b);
        miA[c] = fmaxf(a, fminf(t, b));
        t = v[3][c];
        loB[c] = fminf(t, a);
        hiB[c] = fmaxf(t, b);
        miB[c] = fmaxf(a, fminf(t, b));
    }

    // med9 = med3( max(lows), med3(mids), min(highs) )
    float resA[4], resB[4];
    #pragma unroll
    for (int j = 0; j < 4; ++j) {
        {
            const float mx = fmaxf(fmaxf(loA[j], loA[j + 1]), loA[j + 2]);
            const float mn = fminf(fminf(hiA[j], hiA[j + 1]), hiA[j + 2]);
            const float md = med3f(miA[j], miA[j + 1], miA[j + 2]);
            resA[j] = med3f(mx, md, mn);
        }
        {
            const float mx = fmaxf(fmaxf(loB[j], loB[j + 1]), loB[j + 2]);
            const float mn = fminf(fminf(hiB[j], hiB[j + 1]), hiB[j + 2]);
            const float md = med3f(miB[j], miB[j + 1], miB[j + 2]);
            resB[j] = med3f(mx, md, mn);
        }
    }

    float* __restrict__ dst = out + (unsigned)img * (unsigned)(HO * WO);
    if (oy < HO)
        store_row_nt(dst + (unsigned)(oy * WO + ox), resA, ox);
    if (oy + 1 < HO)
        store_row_nt(dst + (unsigned)((oy + 1) * WO + ox), resB, ox);
}

extern "C" void kernel_launch(void* const* d_in, const int* in_sizes, int n_in,
                              void* d_out, int out_size, void* d_ws, size_t ws_size,
                              hipStream_t stream) {
    const float* x = (const float*)d_in[0];
    float* out = (float*)d_out;
    dim3 grid((WO + TILE_X - 1) / TILE_X,   // 8
              (HO + TILE_Y - 1) / TILE_Y,   // 64
              16 * 3);                      // 48 images
    median3x3_kernel<<<grid, dim3(256), 0, stream>>>(x, out);
}